// YOLOBranch_12086037971081
// MI455X (gfx1250) — compile-verified
//
#include <hip/hip_runtime.h>
#include <hip/hip_bf16.h>
#include <math.h>

#define YB_NA   3
#define YB_NC   80
#define YB_CH   84        // NUM_CLASSES + 4
#define YB_H    64
#define YB_W    64
#define YB_MAXN 64
#define YB_EPS  1e-16f
#define YB_BPB  48        // blocks per batch = NA*H*W/256
#define YB_SELCAP 2000

typedef __attribute__((ext_vector_type(2))) float v2f;
typedef __attribute__((ext_vector_type(8))) float v8f;

#if defined(__gfx1250__) && __has_builtin(__builtin_amdgcn_wmma_f32_16x16x4_f32)
#define YB_HAVE_WMMA4 1
#else
#define YB_HAVE_WMMA4 0
#endif

__constant__ float c_anch[9][2] = {
    {10.f,13.f},{16.f,30.f},{33.f,23.f},{30.f,61.f},{62.f,45.f},
    {59.f,119.f},{116.f,90.f},{156.f,198.f},{373.f,326.f}};

// stable softplus: log(1+e^z) = max(z,0) + log(1+e^-|z|)  (maps to v_exp_f32/v_log_f32)
__device__ __forceinline__ float yb_sp(float z){
    float az = fabsf(z);
    return fmaxf(z, 0.f) + __logf(1.f + __expf(-az));
}
// BCE(sigmoid(z), t) with log clamps at -100:  t=0 -> min(sp(z),100); t=1 -> min(sp(-z),100)
__device__ __forceinline__ float yb_bce0(float z){ return fminf(yb_sp(z), 100.f); }
__device__ __forceinline__ float yb_bce1(float z){ return fminf(yb_sp(-z), 100.f); }

__device__ __forceinline__ float yb_blkredf(float v, float* s){
    int t = threadIdx.x;
    s[t] = v; __syncthreads();
    for (int o = 128; o > 0; o >>= 1){ if (t < o) s[t] += s[t + o]; __syncthreads(); }
    float r = s[0]; __syncthreads();
    return r;
}
__device__ __forceinline__ int yb_blkredi(int v, int* s){
    int t = threadIdx.x;
    s[t] = v; __syncthreads();
    for (int o = 128; o > 0; o >>= 1){ if (t < o) s[t] += s[t + o]; __syncthreads(); }
    int r = s[0]; __syncthreads();
    return r;
}

__global__ void __launch_bounds__(256)
yolo_cell_kernel(const float* __restrict__ raw, const float* __restrict__ labels,
                 const int* __restrict__ p_imgh, const int* __restrict__ p_imgw,
                 float* __restrict__ blkFix, float* __restrict__ blkSall,
                 float* __restrict__ blkSign, int* __restrict__ blkCnt,
                 int B, int N)
{
    __shared__ float s_vrow[YB_MAXN];
    __shared__ int   s_valid[YB_MAXN], s_bn[YB_MAXN], s_ti[YB_MAXN], s_tj[YB_MAXN], s_cls[YB_MAXN];
    __shared__ float s_t0[YB_MAXN], s_t1[YB_MAXN], s_t2[YB_MAXN], s_t3[YB_MAXN];
    __shared__ float s_gx[YB_MAXN], s_gy[YB_MAXN], s_gw[YB_MAXN], s_gh[YB_MAXN];
    __shared__ float s_red[256];
    __shared__ int   s_redi[256];

    const int tid = threadIdx.x;
    const int blk = blockIdx.x;
    const int b   = blk / YB_BPB;
    const int r   = blk - b * YB_BPB;
    const int a   = r >> 4;                      // anchor index 0..2 (16 blocks per anchor plane)
    const int pix = ((r & 15) << 8) | tid;       // j*64+i within the 64x64 plane
    const int jj  = pix >> 6;
    const int ii  = pix & 63;
    const float imgw = (float)p_imgw[0];
    const float imgh = (float)p_imgh[0];
    const int n_lab = (N > YB_MAXN) ? YB_MAXN : N;
    const int CS = YB_H * YB_W;

    // ---- per-label derived data into LDS (threads 0..N-1) ----
    if (tid < n_lab){
        const float* lab = labels + ((size_t)b * N + tid) * 5;
        float l0=lab[0], l1=lab[1], l2=lab[2], l3=lab[3], l4=lab[4];
        float vr = ((l1 + l2 + l3 + l4) > 0.f) ? 1.f : 0.f;
        float tw = l3, th = l4;
        float best = -1.f; int bk = 0;
        #pragma unroll
        for (int k = 0; k < 9; k++){
            float anw = c_anch[k][0] / imgw, anh = c_anch[k][1] / imgh;
            float inter = fminf(tw, anw) * fminf(th, anh);
            float uni   = tw * th + anw * anh - inter;
            float rr    = inter / (uni + YB_EPS);
            if (rr > best){ best = rr; bk = k; }   // argmax, first-max wins
        }
        int bn  = bk % YB_NA;
        int vld = (vr > 0.f) && (bk < YB_NA);
        float tx = l1 * (float)YB_W, ty = l2 * (float)YB_H;
        s_vrow[tid]  = vr;  s_valid[tid] = vld; s_bn[tid] = bn;
        s_ti[tid] = (int)tx; s_tj[tid] = (int)ty;
        s_t0[tid] = tx - floorf(tx); s_t1[tid] = ty - floorf(ty);
        float naw = c_anch[bn][0] / imgw, nah = c_anch[bn][1] / imgh;
        s_t2[tid] = __logf(tw / naw + YB_EPS);
        s_t3[tid] = __logf(th / nah + YB_EPS);
        s_cls[tid] = (int)l0;
        s_gx[tid] = l1; s_gy[tid] = l2; s_gw[tid] = tw; s_gh[tid] = th;
    }
    __syncthreads();

    // ---- per-cell: coalesced channel loads (threads cover 256 consecutive pixels) ----
    const float* cell = raw + ((size_t)b * (YB_NA * YB_CH) + (size_t)a * YB_CH) * CS + pix;
    float z0 = cell[0], z1 = cell[CS], z2 = cell[2*CS], z3 = cell[3*CS];
    float sx = 1.f / (1.f + __expf(-z0));
    float sy = 1.f / (1.f + __expf(-z1));
    float px = fminf(fmaxf((sx + (float)ii) * (1.f/(float)YB_W), 0.f), 1.f);
    float py = fminf(fmaxf((sy + (float)jj) * (1.f/(float)YB_H), 0.f), 1.f);
    float pw = fminf(fmaxf(__expf(z2) * (c_anch[a][0] / imgw), 0.f), 1.f);
    float ph = fminf(fmaxf(__expf(z3) * (c_anch[a][1] / imgh), 0.f), 1.f);

    // best IoU vs all valid_row labels; obj assignment + class bitmask (handles multi-class cells)
    float biou = -1.f; int obj = 0, idx = 0;
    unsigned m0 = 0, m1 = 0, m2 = 0;
    for (int n = 0; n < n_lab; n++){
        if (s_vrow[n] > 0.f){
            float gx=s_gx[n], gy=s_gy[n], gw=s_gw[n], gh=s_gh[n];
            float ix = fminf(px + pw*0.5f, gx + gw*0.5f) - fmaxf(px - pw*0.5f, gx - gw*0.5f);
            float iy = fminf(py + ph*0.5f, gy + gh*0.5f) - fmaxf(py - ph*0.5f, gy - gh*0.5f);
            float inter = fmaxf(ix, 0.f) * fmaxf(iy, 0.f);
            float uni   = pw * ph + gw * gh - inter;
            biou = fmaxf(biou, inter / (uni + YB_EPS));
        }
        if (s_valid[n] && s_bn[n] == a && s_tj[n] == jj && s_ti[n] == ii){
            obj = 1; idx = n;                 // last writer wins (matches scatter)
            int c = s_cls[n];
            if      (c < 32) m0 |= 1u << c;
            else if (c < 64) m1 |= 1u << (c - 32);
            else             m2 |= 1u << (c - 64);
        }
    }

    // class-channel BCE sum (t = one-hot bitmask at obj cells, 0 elsewhere) + sel via max logit
    float bce = 0.f, maxz = -3.0e38f;
    const float* pc = cell + 4 * CS;
    for (int c = 0; c < YB_NC; c++){
        float z = pc[(size_t)c * CS];
        maxz = fmaxf(maxz, z);
        unsigned bit = (c < 32) ? ((m0 >> c) & 1u)
                     : (c < 64) ? ((m1 >> (c - 32)) & 1u)
                                : ((m2 >> (c - 64)) & 1u);
        bce += (obj && bit) ? yb_bce1(z) : yb_bce0(z);
    }
    int sel = (maxz > 0.f) ? 1 : 0;   // sigmoid(max z) > 0.5  <=>  max z > 0

    float fx = 0.f, sa = 0.f, sg = 0.f;
    if (obj){
        float t0 = s_t0[idx], t1 = s_t1[idx];
        float lxy = t0 * yb_bce1(z0) + (1.f - t0) * yb_bce0(z0)
                  + t1 * yb_bce1(z1) + (1.f - t1) * yb_bce0(z1);
        float dw = z2 - s_t2[idx], dh = z3 - s_t3[idx];
        fx = lxy + 0.5f * (dw * dw + dh * dh) + bce;     // loss_xy + 0.5*loss_wh + loss_obj
    } else {
        sa = bce;                                        // candidate noobj contribution
        if (sel && biou > 0.6f) sg = bce;                // removed iff apply_ign for this batch
    }

    float fxr = yb_blkredf(fx, s_red);
    float sar = yb_blkredf(sa, s_red);
    float sgr = yb_blkredf(sg, s_red);
    int   cnr = yb_blkredi(sel, s_redi);
    if (tid == 0){
        blkFix[blk] = fxr; blkSall[blk] = sar; blkSign[blk] = sgr; blkCnt[blk] = cnr;
    }
}

__global__ void __launch_bounds__(32)
yolo_finalize_kernel(const float* __restrict__ labels,
                     const int* __restrict__ p_imgh, const int* __restrict__ p_imgw,
                     const float* __restrict__ blkFix, const float* __restrict__ blkSall,
                     const float* __restrict__ blkSign, const int* __restrict__ blkCnt,
                     float* __restrict__ out, int B, int N)
{
    __shared__ float s_acc[64];
    const int lane = threadIdx.x;       // exactly one wave32, EXEC all ones
    const int NB = B * YB_BPB;
    const float imgw = (float)p_imgw[0], imgh = (float)p_imgh[0];

#if YB_HAVE_WMMA4
    // D = ones(16x4) x B(4x16 data) + C, chained: every D row = running column sums (exact f32).
    v8f acc = {};
    v2f onesA = {1.f, 1.f};
    const int nch = (NB + 63) >> 6;
    for (int ch = 0; ch < nch; ch++){
        int k0 = ch * 64 + lane, k1 = ch * 64 + 32 + lane;
        int i0 = (k0 < NB) ? k0 : 0, i1 = (k1 < NB) ? k1 : 0;
        float v0 = blkFix[i0]; v0 = (k0 < NB) ? v0 : 0.f;   // cndmask, no EXEC change
        float v1 = blkFix[i1]; v1 = (k1 < NB) ? v1 : 0.f;
        v2f bv = {v0, v1};
        acc = __builtin_amdgcn_wmma_f32_16x16x4_f32(false, onesA, false, bv,
                                                    (short)0, acc, false, false);
    }
    s_acc[lane] = acc[0];   // lanes 0..15: D[0][j] = column sums; lanes 16..31 duplicate (row 8)
#else
    float partial = 0.f;
    for (int k = lane; k < NB; k += 32) partial += blkFix[k];
    s_acc[lane] = partial;
#endif

    // per-batch: cnt / S_all / S_ign + 'has any valid label' -> apply_ign decision
    float myno = 0.f;
    for (int b = lane; b < B; b += 32){
        float sall = 0.f, sig = 0.f; int cnt = 0;
        for (int k = 0; k < YB_BPB; k++){
            int id = b * YB_BPB + k;
            sall += blkSall[id]; sig += blkSign[id]; cnt += blkCnt[id];
        }
        int has = 0;
        for (int n = 0; n < N; n++){
            const float* lab = labels + ((size_t)b * N + n) * 5;
            float l1=lab[1], l2=lab[2], l3=lab[3], l4=lab[4];
            if ((l1 + l2 + l3 + l4) > 0.f){
                float best = -1.f; int bk = 0;
                #pragma unroll
                for (int k = 0; k < 9; k++){
                    float anw = c_anch[k][0] / imgw, anh = c_anch[k][1] / imgh;
                    float inter = fminf(l3, anw) * fminf(l4, anh);
                    float uni   = l3 * l4 + anw * anh - inter;
                    float rr    = inter / (uni + YB_EPS);
                    if (rr > best){ best = rr; bk = k; }
                }
                if (bk < YB_NA) has = 1;
            }
        }
        int ap = has && (cnt > 0) && (cnt < YB_SELCAP);
        myno += sall - (ap ? sig : 0.f);
    }
    s_acc[32 + lane] = myno;
    __syncthreads();
    if (lane == 0){
        float fxt = 0.f;
#if YB_HAVE_WMMA4
        for (int q = 0; q < 16; q++) fxt += s_acc[q];
#else
        for (int q = 0; q < 32; q++) fxt += s_acc[q];
#endif
        float no = 0.f;
        for (int q = 0; q < 32; q++) no += s_acc[32 + q];
        out[0] = fxt + no;
    }
}

extern "C" void kernel_launch(void* const* d_in, const int* in_sizes, int n_in,
                              void* d_out, int out_size, void* d_ws, size_t ws_size,
                              hipStream_t stream)
{
    const float* raw    = (const float*)d_in[0];
    const float* labels = (const float*)d_in[1];
    const int*   imgh   = (const int*)d_in[2];
    const int*   imgw   = (const int*)d_in[3];
    int B = in_sizes[0] / (YB_NA * YB_CH * YB_H * YB_W);
    if (B < 1) B = 1;
    int N = in_sizes[1] / (B * 5);
    if (N < 1) N = 1;
    const int NB = B * YB_BPB;

    float* blkFix  = (float*)d_ws;
    float* blkSall = blkFix  + NB;
    float* blkSign = blkSall + NB;
    int*   blkCnt  = (int*)(blkSign + NB);

    yolo_cell_kernel<<<NB, 256, 0, stream>>>(raw, labels, imgh, imgw,
                                             blkFix, blkSall, blkSign, blkCnt, B, N);
    yolo_finalize_kernel<<<1, 32, 0, stream>>>(labels, imgh, imgw,
                                               blkFix, blkSall, blkSign, blkCnt,
                                               (float*)d_out, B, N);
}